// MoreSpecialLSTM_72799695667546
// MI455X (gfx1250) — compile-verified
//
#include <hip/hip_runtime.h>
#include <cstddef>

typedef __attribute__((ext_vector_type(2))) float v2f;
typedef __attribute__((ext_vector_type(8))) float v8f;

#define B_TOT 1024
#define HH    128
#define NROUNDS 10
#define BH    (B_TOT * HH)   // 131072

__device__ __forceinline__ v8f wmma_f32_4(v2f a, v2f b, v8f c) {
    return __builtin_amdgcn_wmma_f32_16x16x4_f32(false, a, false, b, (short)0, c, false, false);
}

__device__ __forceinline__ float sigf(float x) { return 1.0f / (1.0f + __expf(-x)); }

// ---------------------------------------------------------------------------
// Generic C[M,N] = act(A[M,K] @ W[N,K]^T + bias[N]), fp32 WMMA 16x16x4.
// grid = (Mrows/16, N/128), block = 256 (8 waves, one 16-col tile each).
// K must be a multiple of 128.
// ---------------------------------------------------------------------------
__global__ __launch_bounds__(256)
void gemm_bias_relu(const float* __restrict__ A, const float* __restrict__ W,
                    const float* __restrict__ bias, float* __restrict__ C,
                    int N, int K, int do_relu)
{
    __shared__ float Alds[16 * 128];
    const int tid  = threadIdx.x;
    const int wave = tid >> 5, lane = tid & 31;
    const int lh   = lane >> 4, l16 = lane & 15;
    const int row0 = blockIdx.x * 16;
    const int n    = blockIdx.y * 128 + wave * 16 + l16;

    v8f acc = {};
    for (int kc = 0; kc < K; kc += 128) {
        for (int idx = tid; idx < 16 * 128; idx += 256) {
            int m = idx >> 7, k = idx & 127;
            Alds[idx] = A[(size_t)(row0 + m) * K + kc + k];
        }
        __syncthreads();
        #pragma unroll 8
        for (int kk = 0; kk < 128; kk += 4) {
            const int kb = kk + lh * 2;
            v2f a; a.x = Alds[(l16 << 7) + kb]; a.y = Alds[(l16 << 7) + kb + 1];
            const float* p = W + (size_t)n * K + kc + kb;
            v2f b; b.x = p[0]; b.y = p[1];
            acc = wmma_f32_4(a, b, acc);
        }
        __syncthreads();
    }
    const float bn = bias[n];
    #pragma unroll
    for (int v = 0; v < 8; ++v) {
        float val = acc[v] + bn;
        if (do_relu) val = fmaxf(val, 0.0f);
        C[(size_t)(row0 + v + lh * 8) * N + n] = val;
    }
}

// ---------------------------------------------------------------------------
// Fused LSTM recurrence + attention. Each block owns 16 batch rows and runs
// the full (r, hist, t) schedule; h/c live in LDS for the whole kernel.
// ---------------------------------------------------------------------------
struct AttnScratch {
    float Qs[16 * 128];
    float Ks[16 * 128];
    float T1[16 * 64];
    float Vl[NROUNDS * 32];   // [slot][row*2 + d]  (log-softmax V)
    float sc[16 * NROUNDS];   // raw attention scores per row
};
union Scratch {
    float Alds[16 * 256];
    AttnScratch at;
};

// One LSTM layer step for this block's 16 rows.
// inp: row m at inp + m*inp_stride (128 floats). Gg: per-block global gate slab.
__device__ __forceinline__ void lstm_step_dev(
    const float* inp, int inp_stride,
    const float* __restrict__ Wih, const float* __restrict__ Whh,
    const float* __restrict__ bih, const float* __restrict__ bhh,
    float* Hs, float* Cs, float* Alds, float* __restrict__ Gg, int tid)
{
    // Stage A = [inp | h] : 16 x 256
    for (int idx = tid; idx < 16 * 256; idx += 256) {
        int m = idx >> 8, k = idx & 255;
        Alds[idx] = (k < 128) ? inp[m * inp_stride + k] : Hs[(m << 7) + (k - 128)];
    }
    __syncthreads();

    const int wave = tid >> 5, lane = tid & 31;
    const int lh = lane >> 4, l16 = lane & 15;
    for (int tile = 0; tile < 4; ++tile) {
        const int n = wave * 64 + tile * 16 + l16;   // gate column 0..511
        v8f acc = {};
        #pragma unroll 8
        for (int kk = 0; kk < 256; kk += 4) {
            const int kb = kk + lh * 2;
            v2f a; a.x = Alds[(l16 << 8) + kb]; a.y = Alds[(l16 << 8) + kb + 1];
            const float* p = (kk < 128) ? (Wih + (size_t)n * 128 + kb)
                                        : (Whh + (size_t)n * 128 + (kb - 128));
            v2f b; b.x = p[0]; b.y = p[1];
            acc = wmma_f32_4(a, b, acc);
        }
        const float bsum = bih[n] + bhh[n];
        #pragma unroll
        for (int v = 0; v < 8; ++v)
            Gg[((v + lh * 8) << 9) + n] = acc[v] + bsum;
    }
    __threadfence_block();
    __syncthreads();

    // Pointwise gate update (i,f,g,o -> c,h)
    for (int idx = tid; idx < 16 * 128; idx += 256) {
        const int m = idx >> 7, j = idx & 127;
        const float* g = Gg + (m << 9);
        const float ig = sigf(g[j]);
        const float fg = sigf(g[128 + j]);
        const float gg = tanhf(g[256 + j]);
        const float og = sigf(g[384 + j]);
        const float cn = fg * Cs[idx] + ig * gg;
        Cs[idx] = cn;
        Hs[idx] = og * tanhf(cn);
    }
    __syncthreads();
}

// Dst[16,N] = act(Asrc[16,128] @ W[N,128]^T + bias). N in {64,128}. All threads call.
__device__ __forceinline__ void proj16_dev(
    const float* Asrc, const float* __restrict__ W, const float* __restrict__ bias,
    float* Dst, int N, bool do_relu, int tid)
{
    const int wave = tid >> 5, lane = tid & 31;
    const int lh = lane >> 4, l16 = lane & 15;
    if (wave * 16 < N) {
        const int n = wave * 16 + l16;
        v8f acc = {};
        #pragma unroll 8
        for (int kk = 0; kk < 128; kk += 4) {
            const int kb = kk + lh * 2;
            v2f a; a.x = Asrc[(l16 << 7) + kb]; a.y = Asrc[(l16 << 7) + kb + 1];
            const float* p = W + (size_t)n * 128 + kb;
            v2f b; b.x = p[0]; b.y = p[1];
            acc = wmma_f32_4(a, b, acc);
        }
        const float bn = bias[n];
        #pragma unroll
        for (int v = 0; v < 8; ++v) {
            float val = acc[v] + bn;
            if (do_relu) val = fmaxf(val, 0.0f);
            Dst[(v + lh * 8) * N + n] = val;
        }
    }
    __syncthreads();
}

__global__ __launch_bounds__(256)
void lstm_attn_kernel(const float* __restrict__ lstm_in,   // [B,10,128]
                      const float* __restrict__ game,      // [2*B*128] flat h0
                      const float* __restrict__ user,      // [2*B*128] flat c0
                      const float* __restrict__ Wih, const float* __restrict__ Whh,
                      const float* __restrict__ bih, const float* __restrict__ bhh,
                      const float* __restrict__ Wq,  const float* __restrict__ bq,
                      const float* __restrict__ Wk,  const float* __restrict__ bk,
                      const float* __restrict__ Wo1, const float* __restrict__ bo1,
                      const float* __restrict__ Wo2, const float* __restrict__ bo2,
                      float* __restrict__ Mbuf,   // [10][B][128]
                      float* __restrict__ Gbuf,   // [gridDim.x][16][512]
                      float* __restrict__ out,    // [B,10,2]
                      float* __restrict__ att,    // [B,10,10]
                      float* __restrict__ hout,   // [2,B,128] flat
                      float* __restrict__ cout)   // [2,B,128] flat
{
    __shared__ Scratch S;
    __shared__ float Hs[2][16 * 128];
    __shared__ float Cs[2][16 * 128];

    const int tid = threadIdx.x;
    const int bb  = blockIdx.x * 16;            // first batch row of this block
    float* Gg = Gbuf + (size_t)blockIdx.x * 16 * 512;

    // init h, c (flat reinterpretation (2, B, 128) of game/user vectors)
    for (int l = 0; l < 2; ++l)
        for (int idx = tid; idx < 16 * 128; idx += 256) {
            Hs[l][idx] = game[(size_t)l * BH + bb * HH + idx];
            Cs[l][idx] = user[(size_t)l * BH + bb * HH + idx];
        }
    __syncthreads();

    for (int r = 0; r < NROUNDS; ++r) {
        for (int hist = 0; hist <= r; ++hist) {
            const int t0 = (hist > 1) ? hist : 0;
            for (int t = t0; t <= r; ++t) {
                lstm_step_dev(lstm_in + (size_t)bb * (NROUNDS * HH) + t * HH, NROUNDS * HH,
                              Wih, Whh, bih, bhh, Hs[0], Cs[0], S.Alds, Gg, tid);
                lstm_step_dev(Hs[0], HH,
                              Wih + 512 * 128, Whh + 512 * 128, bih + 512, bhh + 512,
                              Hs[1], Cs[1], S.Alds, Gg, tid);
            }
            // M[hist] = h of top layer
            float* dst = Mbuf + (size_t)hist * BH + (size_t)bb * HH;
            for (int idx = tid; idx < 16 * 128; idx += 256) dst[idx] = Hs[1][idx];
            __threadfence_block();
            __syncthreads();
        }

        // ---- attention for round r (per batch row, fully block-local) ----
        AttnScratch* AT = &S.at;
        proj16_dev(Mbuf + (size_t)r * BH + (size_t)bb * HH, Wq, bq, AT->Qs, 128, false, tid);
        for (int k = 0; k <= r; ++k) {
            const float* Mk = Mbuf + (size_t)k * BH + (size_t)bb * HH;
            proj16_dev(Mk, Wk, bk, AT->Ks, 128, false, tid);
            if (tid < 16) {
                float s = 0.0f;
                const float* q  = AT->Qs + tid * 128;
                const float* kv = AT->Ks + tid * 128;
                for (int j = 0; j < 128; ++j) s += q[j] * kv[j];
                AT->sc[tid * NROUNDS + k] = s * 0.08838834764831845f;  // 1/sqrt(128)
            }
            proj16_dev(Mk, Wo1, bo1, AT->T1, 64, true, tid);
            if (tid < 16) {
                float l0 = bo2[0], l1 = bo2[1];
                const float* t1 = AT->T1 + tid * 64;
                for (int j = 0; j < 64; ++j) { l0 += t1[j] * Wo2[j]; l1 += t1[j] * Wo2[64 + j]; }
                const float mx  = fmaxf(l0, l1);
                const float lse = mx + logf(__expf(l0 - mx) + __expf(l1 - mx));
                AT->Vl[k * 32 + tid * 2 + 0] = l0 - lse;
                AT->Vl[k * 32 + tid * 2 + 1] = l1 - lse;
            }
            __syncthreads();
        }
        if (tid < 16) {
            const int b = bb + tid;
            float mx = -1e30f;
            for (int k = 0; k <= r; ++k) mx = fmaxf(mx, AT->sc[tid * NROUNDS + k]);
            float p[NROUNDS]; float s = 0.0f;
            for (int k = 0; k <= r; ++k) { p[k] = __expf(AT->sc[tid * NROUNDS + k] - mx); s += p[k]; }
            const float inv = 1.0f / s;
            float o0 = 0.0f, o1 = 0.0f;
            for (int k = 0; k <= r; ++k) {
                const float pk = p[k] * inv;
                att[(size_t)b * 100 + r * 10 + k] = pk;
                o0 += pk * AT->Vl[k * 32 + tid * 2 + 0];
                o1 += pk * AT->Vl[k * 32 + tid * 2 + 1];
            }
            for (int k = r + 1; k < NROUNDS; ++k) att[(size_t)b * 100 + r * 10 + k] = -1.0f;
            out[(size_t)b * 20 + r * 2 + 0] = o0;
            out[(size_t)b * 20 + r * 2 + 1] = o1;
        }
        __syncthreads();
    }

    // final h, c (flat (2,B,128) == reference reshape to (B,2,128))
    for (int l = 0; l < 2; ++l)
        for (int idx = tid; idx < 16 * 128; idx += 256) {
            hout[(size_t)l * BH + bb * HH + idx] = Hs[l][idx];
            cout[(size_t)l * BH + bb * HH + idx] = Cs[l][idx];
        }
}

// ---------------------------------------------------------------------------
extern "C" void kernel_launch(void* const* d_in, const int* in_sizes, int n_in,
                              void* d_out, int out_size, void* d_ws, size_t ws_size,
                              hipStream_t stream) {
    (void)in_sizes; (void)n_in; (void)out_size; (void)ws_size;
    const float* input_vec = (const float*)d_in[0];   // [1024,10,512]
    const float* game      = (const float*)d_in[1];   // [1024,2,128]
    const float* user      = (const float*)d_in[2];
    const float* W_in1     = (const float*)d_in[3];   // [1024,512]
    const float* b_in1     = (const float*)d_in[4];
    const float* W_in2     = (const float*)d_in[5];   // [128,1024]
    const float* b_in2     = (const float*)d_in[6];
    const float* Wq        = (const float*)d_in[7];
    const float* bq        = (const float*)d_in[8];
    const float* Wk        = (const float*)d_in[9];
    const float* bk        = (const float*)d_in[10];
    const float* Wih       = (const float*)d_in[11];  // [2,512,128]
    const float* Whh       = (const float*)d_in[12];
    const float* bih       = (const float*)d_in[13];  // [2,512]
    const float* bhh       = (const float*)d_in[14];
    const float* Wo1       = (const float*)d_in[15];  // [64,128]
    const float* bo1       = (const float*)d_in[16];
    const float* Wo2       = (const float*)d_in[17];  // [2,64]
    const float* bo2       = (const float*)d_in[18];

    float* ws      = (float*)d_ws;
    float* T0      = ws;                       // [1024,1024] chunk buffer
    float* lstm_in = T0 + 1024 * 1024;         // [10240,128]
    float* Mbuf    = lstm_in + 10240 * 128;    // [10,1024,128]
    float* Gbuf    = Mbuf + 10 * 1024 * 128;   // [64,16,512]

    float* out  = (float*)d_out;               // [1024,10,2]
    float* attn = out + 1024 * 10 * 2;         // [1024,10,10]
    float* hout = attn + 1024 * 10 * 10;       // [2,1024,128]
    float* cout = hout + 2 * 1024 * 128;

    // Input MLP: lstm_in = relu(relu(X @ W1^T + b1) @ W2^T + b2), row-chunked.
    for (int c = 0; c < 10; ++c) {
        gemm_bias_relu<<<dim3(64, 8), 256, 0, stream>>>(
            input_vec + (size_t)c * 1024 * 512, W_in1, b_in1, T0, 1024, 512, 1);
        gemm_bias_relu<<<dim3(64, 1), 256, 0, stream>>>(
            T0, W_in2, b_in2, lstm_in + (size_t)c * 1024 * 128, 128, 1024, 1);
    }

    // Fused LSTM recurrence + attention: 64 blocks x 16 batch rows.
    lstm_attn_kernel<<<64, 256, 0, stream>>>(
        lstm_in, game, user, Wih, Whh, bih, bhh,
        Wq, bq, Wk, bk, Wo1, bo1, Wo2, bo2,
        Mbuf, Gbuf, out, attn, hout, cout);
}